// SimpleGNN_6571299963443
// MI455X (gfx1250) — compile-verified
//
#include <hip/hip_runtime.h>
#include <hip/hip_bf16.h>

typedef float v2f __attribute__((ext_vector_type(2)));
typedef float v8f __attribute__((ext_vector_type(8)));

#define DIM 128

// ---------------------------------------------------------------------------
// Elementwise / degree kernels
// ---------------------------------------------------------------------------
__global__ void zero_f(float* p, int n) {
    int i = blockIdx.x * blockDim.x + threadIdx.x;
    if (i < n) p[i] = 0.0f;
}

__global__ void deg_accum(const int* __restrict__ ei, int E, float* deg) {
    int e = blockIdx.x * blockDim.x + threadIdx.x;
    if (e < E) unsafeAtomicAdd(&deg[ei[E + e]], 1.0f);
}

__global__ void deg_to_dinv(float* d, int n) {
    int i = blockIdx.x * blockDim.x + threadIdx.x;
    if (i < n) d[i] = rsqrtf(d[i] + 1.0f);
}

// agg[n][c] = h[n][c] * dinv[n]^2   (self-loop term)
__global__ void init_agg(const float* __restrict__ h, const float* __restrict__ dinv,
                         float* __restrict__ agg, int total) {
    int i = blockIdx.x * blockDim.x + threadIdx.x;
    if (i < total) {
        float di = dinv[i >> 7];
        agg[i] = h[i] * di * di;
    }
}

__global__ void bias_relu(const float* __restrict__ in, const float* __restrict__ b,
                          float* __restrict__ out, int total) {
    int i = blockIdx.x * blockDim.x + threadIdx.x;
    if (i < total) out[i] = fmaxf(in[i] + b[i & (DIM - 1)], 0.0f);
}

// ---------------------------------------------------------------------------
// Edge scatter: one wave32 per edge, float4 per lane (4*32 = 128 features)
// agg[dst] += h[src] * dinv[src] * dinv[dst]
// ---------------------------------------------------------------------------
__global__ __launch_bounds__(256) void scatter_edges(const float* __restrict__ H,
                                                     const float* __restrict__ dinv,
                                                     const int* __restrict__ ei, int E,
                                                     float* __restrict__ AGG) {
    int e = blockIdx.x * (blockDim.x >> 5) + (threadIdx.x >> 5);
    if (e >= E) return;
    int lane = threadIdx.x & 31;
    int s = ei[e];
    int d = ei[E + e];
    float norm = dinv[s] * dinv[d];
    const float4 h4 = *(const float4*)(H + (size_t)s * DIM + lane * 4);
    float* dst = AGG + (size_t)d * DIM + lane * 4;
    unsafeAtomicAdd(dst + 0, h4.x * norm);
    unsafeAtomicAdd(dst + 1, h4.y * norm);
    unsafeAtomicAdd(dst + 2, h4.z * norm);
    unsafeAtomicAdd(dst + 3, h4.w * norm);
}

// ---------------------------------------------------------------------------
// GEMM  Y[M x 128] = X[M x 128] @ W[128 x 128]  via V_WMMA_F32_16X16X4_F32.
// Block = 256 threads = 8 waves; each wave owns one 16-row strip x 128 cols
// (8 C tiles, K in steps of 4).
//
// Staging (once per block):
//   1) async-copy W (64 KB, row-major) into LDS  (global_load_async_to_lds_b128)
//   2) repack into WMMA B-fragment layout:
//        Bfrag[ks][nt][lane] = { W[4ks+kHi][n], W[4ks+kHi+1][n] }
//      with kHi=(lane>>4)*2, n=nt*16+(lane&15)  -> inner loop is exactly one
//      lane-contiguous ds_load_b64 per WMMA (pairs fuse to ds_load_2addr_b64).
// LDS layout: Bfrag in the LOW 64 KB so every inner-loop DS load uses a single
// base VGPR + 16-bit immediate offset; Wrow staging buffer in the high 64 KB.
// ---------------------------------------------------------------------------
__global__ __launch_bounds__(256) void gemm_wmma_f32(const float* __restrict__ X,
                                                     const float* __restrict__ W,
                                                     float* __restrict__ Y, int M) {
    extern __shared__ float smem[];
    v2f*   Bfrag = (v2f*)smem;                // [32][8][32] v2f = 64 KB @ offset 0
    float* Wrow  = smem + DIM * DIM;          // [128][128] row-major, upper 64 KB

    const int t    = threadIdx.x;             // 0..255
    const int lane = t & 31;
    const int wave = t >> 5;                  // 0..7

    // --- 1) async-stage W into LDS: 256 threads x 16 B x 16 iters = 64 KB ---
    {
        unsigned ldsBase = (unsigned)(size_t)(&Wrow[0]) + (unsigned)t * 16u;
        unsigned gOff    = (unsigned)t * 16u;
#pragma unroll
        for (int i = 0; i < 16; ++i) {
            asm volatile("global_load_async_to_lds_b128 %0, %1, %2 offset:0"
                         :
                         : "v"(ldsBase + (unsigned)i * 4096u),
                           "v"(gOff + (unsigned)i * 4096u),
                           "s"(W)
                         : "memory");
        }
        asm volatile("s_wait_asynccnt 0" ::: "memory");
    }
    __syncthreads();

    // --- 2) repack into B-fragment layout (wave w covers ks = 4w..4w+3) ---
    {
        const int kHi = (lane >> 4) * 2;
        const int c   = lane & 15;
        for (int i = 0; i < 4; ++i) {
            const int ks = wave * 4 + i;
            const int k  = ks * 4 + kHi;
#pragma unroll
            for (int nt = 0; nt < 8; ++nt) {
                const int n = nt * 16 + c;
                v2f b;
                b.x = Wrow[k * DIM + n];
                b.y = Wrow[(k + 1) * DIM + n];
                Bfrag[(ks * 8 + nt) * 32 + lane] = b;
            }
        }
    }
    __syncthreads();

    // --- 3) WMMA main loop: one 16-row strip per wave ---
    const int strips = (M + 15) >> 4;
    const int strip  = blockIdx.x * 8 + wave;
    if (strip < strips) {
        const int row0 = strip * 16;
        const int aRow = row0 + (lane & 15);       // A: 16x4 f32 lane layout
        const int kHi  = (lane >> 4) * 2;          // 0 or 2
        const int bCol = lane & 15;
        const float* xrow = X + (size_t)aRow * DIM + kHi;
        const v2f* bbase = Bfrag + lane;           // + imm offsets ks*2048+nt*256 B

        v8f acc[8] = {};
        for (int ks = 0; ks < 32; ++ks) {
            v2f a = *(const v2f*)(xrow + ks * 4);
            // preload all 8 B fragments -> DS loads issue as a clause and are
            // drained with staggered waits, overlapping LDS latency with WMMA
            v2f bf[8];
#pragma unroll
            for (int nt = 0; nt < 8; ++nt) bf[nt] = bbase[(ks * 8 + nt) * 32];
#pragma unroll
            for (int nt = 0; nt < 8; ++nt) {
                acc[nt] = __builtin_amdgcn_wmma_f32_16x16x4_f32(
                    false, a, false, bf[nt], (short)0, acc[nt], false, false);
            }
        }
        // C/D layout: VGPR j -> row0 + j (lanes 0-15) / row0 + j + 8 (lanes 16-31)
        const int cRow = row0 + (lane >> 4) * 8;
#pragma unroll
        for (int nt = 0; nt < 8; ++nt) {
#pragma unroll
            for (int j = 0; j < 8; ++j) {
                Y[(size_t)(cRow + j) * DIM + nt * 16 + bCol] = acc[nt][j];
            }
        }
    }
}

// ---------------------------------------------------------------------------
// Final: out[n] = dot(relu(agg[n] + b2), Wl) + bl   (one wave per node)
// ---------------------------------------------------------------------------
__global__ __launch_bounds__(256) void final_dot(const float* __restrict__ AGG,
                                                 const float* __restrict__ b2,
                                                 const float* __restrict__ Wl,
                                                 const float* __restrict__ bl,
                                                 float* __restrict__ out, int N) {
    int n = blockIdx.x * (blockDim.x >> 5) + (threadIdx.x >> 5);
    if (n >= N) return;
    int lane = threadIdx.x & 31;
    const float4 a = *(const float4*)(AGG + (size_t)n * DIM + lane * 4);
    const float4 b = *(const float4*)(b2 + lane * 4);
    const float4 w = *(const float4*)(Wl + lane * 4);
    float p = fmaxf(a.x + b.x, 0.0f) * w.x + fmaxf(a.y + b.y, 0.0f) * w.y +
              fmaxf(a.z + b.z, 0.0f) * w.z + fmaxf(a.w + b.w, 0.0f) * w.w;
#pragma unroll
    for (int off = 16; off > 0; off >>= 1) p += __shfl_xor(p, off, 32);
    if (lane == 0) out[n] = p + bl[0];
}

// ---------------------------------------------------------------------------
// Launch
// ---------------------------------------------------------------------------
static inline int ceil_div(int a, int b) { return (a + b - 1) / b; }

extern "C" void kernel_launch(void* const* d_in, const int* in_sizes, int n_in,
                              void* d_out, int out_size, void* d_ws, size_t ws_size,
                              hipStream_t stream) {
    const float* x  = (const float*)d_in[0];
    const int*   ei = (const int*)d_in[1];
    const float* W1 = (const float*)d_in[2];
    const float* b1 = (const float*)d_in[3];
    const float* W2 = (const float*)d_in[4];
    const float* b2 = (const float*)d_in[5];
    const float* Wl = (const float*)d_in[6];
    const float* bl = (const float*)d_in[7];
    float* out = (float*)d_out;

    const int N = in_sizes[0] / DIM;   // 50000
    const int E = in_sizes[1] / 2;     // 800000
    const int total = N * DIM;

    // workspace: dinv | bufA | bufB  (~51.5 MB)
    float* dinv = (float*)d_ws;
    size_t npad = ((size_t)N + 511) & ~(size_t)511;
    float* bufA = dinv + npad;
    float* bufB = bufA + (size_t)N * DIM;

    const int strips     = ceil_div(N, 16);
    const int gemmBlocks = ceil_div(strips, 8);
    const int ewBlocks   = ceil_div(total, 256);
    const int edgeWave   = ceil_div(E, 8);          // 8 waves/block in scatter
    const size_t gemmLds = (size_t)2 * DIM * DIM * sizeof(float);  // 128 KB

    // degree -> dinv
    zero_f<<<ceil_div(N, 256), 256, 0, stream>>>(dinv, N);
    deg_accum<<<ceil_div(E, 256), 256, 0, stream>>>(ei, E, dinv);
    deg_to_dinv<<<ceil_div(N, 256), 256, 0, stream>>>(dinv, N);

    // layer 1
    gemm_wmma_f32<<<gemmBlocks, 256, gemmLds, stream>>>(x, W1, bufA, N);
    init_agg<<<ewBlocks, 256, 0, stream>>>(bufA, dinv, bufB, total);
    scatter_edges<<<edgeWave, 256, 0, stream>>>(bufA, dinv, ei, E, bufB);
    bias_relu<<<ewBlocks, 256, 0, stream>>>(bufB, b1, bufA, total);

    // layer 2
    gemm_wmma_f32<<<gemmBlocks, 256, gemmLds, stream>>>(bufA, W2, bufB, N);
    init_agg<<<ewBlocks, 256, 0, stream>>>(bufB, dinv, bufA, total);
    scatter_edges<<<edgeWave, 256, 0, stream>>>(bufB, dinv, ei, E, bufA);

    // head
    final_dot<<<ceil_div(N, 8), 256, 0, stream>>>(bufA, b2, Wl, bl, out, N);
}